// SparseDepthwiseConv2d_7284264534725
// MI455X (gfx1250) — compile-verified
//
#include <hip/hip_runtime.h>

// Depthwise 3x3 conv, stride 1, pad 1. N=16, C=256, H=W=128, fp32.
// Memory-bound (~537 MB moved -> ~23 us at 23.3 TB/s). Strategy:
// async global->LDS staging (gfx1250 ASYNC path) + register sliding-window
// 3x3 stencil from LDS, non-temporal coalesced fp32 stores.

#define DW_H 128
#define DW_W 128
#define DW_C 256
#define DW_ROWS 32            // output rows per workgroup
#define DW_LROWS 34           // staged rows (with vertical halo)
#define DW_LSTR 132           // floats per LDS row (128 data + 4 zero pad)
#define DW_LPAD 4             // leading pad floats (zeroed; serves row0 col-1)
#define DW_LDS_FLOATS (DW_LPAD + DW_LROWS * DW_LSTR)   // 4492 floats ~ 17.9 KB

typedef int v4i_t __attribute__((ext_vector_type(4)));
typedef __attribute__((address_space(1))) v4i_t* as1_v4i;
typedef __attribute__((address_space(3))) v4i_t* as3_v4i;

__device__ __forceinline__ void async_copy16(float* lds_dst, const float* gsrc) {
#if __has_builtin(__builtin_amdgcn_global_load_async_to_lds_b128)
    __builtin_amdgcn_global_load_async_to_lds_b128(
        (as1_v4i)gsrc, (as3_v4i)lds_dst, /*offset=*/0, /*cpol=*/0);
#else
    // Fallback: synchronous copy through registers (still correct).
    *(float4*)lds_dst = *(const float4*)gsrc;
#endif
}

__device__ __forceinline__ void wait_async_done() {
#if __has_builtin(__builtin_amdgcn_s_wait_asynccnt)
    __builtin_amdgcn_s_wait_asynccnt(0);
#else
    asm volatile("s_wait_asynccnt 0" ::: "memory");
#endif
}

__global__ __launch_bounds__(256)
void dwconv3x3_async_kernel(const float* __restrict__ x,
                            const float* __restrict__ w,
                            const float* __restrict__ bias,
                            float* __restrict__ y) {
    __shared__ float smem[DW_LDS_FLOATS];

    const int tid = threadIdx.x;
    const int rb  = blockIdx.x;                 // 0..3 (row block)
    const int p   = blockIdx.y;                 // plane = n*C + c, 0..4095
    const int c   = p & (DW_C - 1);

    const float* __restrict__ xp = x + (size_t)p * (DW_H * DW_W);
    float*       __restrict__ yp = y + (size_t)p * (DW_H * DW_W);
    const int gRow0 = rb * DW_ROWS - 1;         // global row of LDS row 0

    // ---- zero pads: 4 leading floats + 4 trailing floats per row (140 total)
    if (tid < DW_LPAD) smem[tid] = 0.0f;
    {
        int i = tid - DW_LPAD;                  // threads 4..139 handle row pads
        if (i >= 0 && i < DW_LROWS * 4) {
            int r = i >> 2, j = i & 3;
            smem[DW_LPAD + r * DW_LSTR + 128 + j] = 0.0f;
        }
    }
    // ---- zero out-of-range halo rows (only LDS rows 0 and 33 can be OOB)
    if (tid < 128) {
        if (gRow0 < 0)
            smem[DW_LPAD + 0 * DW_LSTR + tid] = 0.0f;
        if (gRow0 + (DW_LROWS - 1) >= DW_H)
            smem[DW_LPAD + (DW_LROWS - 1) * DW_LSTR + tid] = 0.0f;
    }

    // ---- async stage valid rows: 34 rows x 32 chunks of 16B = 1088 chunks
    for (int k = tid; k < DW_LROWS * 32; k += 256) {
        int r  = k >> 5;
        int ch = k & 31;
        int gr = gRow0 + r;
        if (gr >= 0 && gr < DW_H) {
            async_copy16(&smem[DW_LPAD + r * DW_LSTR + ch * 4],
                         xp + gr * DW_W + ch * 4);
        }
    }
    wait_async_done();
    __syncthreads();

    // ---- per-channel taps (uniform per workgroup -> scalar loads)
    const float* __restrict__ wc = w + c * 9;
    const float w00 = wc[0], w01 = wc[1], w02 = wc[2];
    const float w10 = wc[3], w11 = wc[4], w12 = wc[5];
    const float w20 = wc[6], w21 = wc[7], w22 = wc[8];
    const float bb  = bias[c];

    // ---- compute: thread = (column, half); 16 rows per thread,
    //      3x3 register sliding window, 3 new LDS reads per output row.
    const int col  = tid & 127;
    const int half = tid >> 7;

    int idx = DW_LPAD + (half * 16) * DW_LSTR + col;   // LDS row half*16
    float l0 = smem[idx - 1], m0 = smem[idx], r0 = smem[idx + 1];
    idx += DW_LSTR;
    float l1 = smem[idx - 1], m1 = smem[idx], r1 = smem[idx + 1];

    float* __restrict__ yrow = yp + (rb * DW_ROWS + half * 16) * DW_W + col;

#pragma unroll
    for (int o = 0; o < 16; ++o) {
        idx += DW_LSTR;
        float l2 = smem[idx - 1], m2 = smem[idx], r2 = smem[idx + 1];
        float acc = bb;
        acc = fmaf(w00, l0, acc);
        acc = fmaf(w01, m0, acc);
        acc = fmaf(w02, r0, acc);
        acc = fmaf(w10, l1, acc);
        acc = fmaf(w11, m1, acc);
        acc = fmaf(w12, r1, acc);
        acc = fmaf(w20, l2, acc);
        acc = fmaf(w21, m2, acc);
        acc = fmaf(w22, r2, acc);
        // Output is write-once/never re-read: stream past L2 (TH_STORE_NT)
        __builtin_nontemporal_store(acc, &yrow[o * DW_W]);
        l0 = l1; m0 = m1; r0 = r1;
        l1 = l2; m1 = m2; r1 = r2;
    }
}

extern "C" void kernel_launch(void* const* d_in, const int* in_sizes, int n_in,
                              void* d_out, int out_size, void* d_ws, size_t ws_size,
                              hipStream_t stream) {
    (void)n_in; (void)out_size; (void)d_ws; (void)ws_size;
    const float* x    = (const float*)d_in[0];   // (N, C, H, W)
    const float* w    = (const float*)d_in[1];   // (C, 1, 3, 3)
    const float* bias = (const float*)d_in[2];   // (C,)
    float* y = (float*)d_out;

    const int planes = in_sizes[0] / (DW_H * DW_W);      // N*C = 4096
    dim3 grid(DW_H / DW_ROWS, (unsigned)planes);          // 4 x 4096
    dwconv3x3_async_kernel<<<grid, 256, 0, stream>>>(x, w, bias, y);
}